// LearnableDemosaick_16509854286238
// MI455X (gfx1250) — compile-verified
//
#include <hip/hip_runtime.h>
#include <hip/hip_bf16.h>

typedef __attribute__((ext_vector_type(2))) float v2f;
typedef __attribute__((ext_vector_type(8))) float v8f;

#define TILE_W 64
#define TILE_H 16
#define HALO   2
#define LDSW   (TILE_W + 2 * HALO)   // 68
#define LDSH   (TILE_H + 2 * HALO)   // 20
#define NT     256                   // 8 waves (wave32)

// gfx1250 async global->LDS path (GLOBAL_LOAD_ASYNC_TO_LDS_B32, ASYNCcnt),
// gated on toolchain support so the kernel always compiles.
#if defined(__has_builtin)
#if __has_builtin(__builtin_amdgcn_global_load_async_to_lds_b32) && \
    __has_builtin(__builtin_amdgcn_s_wait_asynccnt)
#define USE_ASYNC_LDS 1
#endif
#endif
#ifndef USE_ASYNC_LDS
#define USE_ASYNC_LDS 0
#endif

#if USE_ASYNC_LDS
typedef __attribute__((address_space(1))) int gint_as1;
typedef __attribute__((address_space(3))) int lint_as3;
#endif

// Fill a (LDSH x LDSW) zero-padded halo tile from one image plane.
// In-bounds elements stream via async global->LDS copy (no VGPR staging);
// out-of-image halo lanes write 0.0f through DS (SAME padding).
__device__ __forceinline__ void tile_fill(float* tile, const float* __restrict__ img,
                                          int x0, int y0, int H, int W, int tid)
{
    for (int i = tid; i < LDSH * LDSW; i += NT) {
        int ly = i / LDSW;
        int lx = i - ly * LDSW;
        int gy = y0 + ly - HALO;
        int gx = x0 + lx - HALO;
        bool inb = (gy >= 0) & (gy < H) & (gx >= 0) & (gx < W);
#if USE_ASYNC_LDS
        if (inb) {
            const float* g = img + (size_t)gy * W + gx;
            __builtin_amdgcn_global_load_async_to_lds_b32(
                (gint_as1*)(uintptr_t)g,
                (lint_as3*)(uint32_t)(uintptr_t)(&tile[i]),
                0, 0);
        } else {
            tile[i] = 0.0f;
        }
#else
        float v = 0.0f;
        if (inb) v = img[(size_t)gy * W + gx];
        tile[i] = v;
#endif
    }
#if USE_ASYNC_LDS
    __builtin_amdgcn_s_wait_asynccnt(0);
#endif
}

// ---------------------------------------------------------------------------
// Pass 1: 16-filter bank (8 sel + 8 green) via V_WMMA_F32_16X16X4_F32,
// softmax-weighted green interpolation, emit green + chroma-diff planes.
//
// WMMA mapping per 16-pixel group (one wave):
//   A (16x4, f32): M = filter index (0..7 sel, 8..15 green), K = tap chunk
//                  lane&15 = M ; lane>>4 selects K pair {4kk,4kk+1} vs {4kk+2,4kk+3}
//   B (4x16, f32): N = pixel column (lane&15), same K mapping as A
//   C/D (16x16,f32): lanes 0-15 hold sel[m] (VGPR m), lanes 16-31 hold green[m]
//                    for the SAME pixel N = lane&15.
// 25 taps padded to 28 -> 7 chained WMMAs; padded K entries have A == 0.
// ---------------------------------------------------------------------------
__global__ __launch_bounds__(NT) void demosaic_pass1(
    const float* __restrict__ mosaick,
    const float* __restrict__ sel_filts,     // [8][5][5]
    const float* __restrict__ green_filts,   // [8][5][5]
    float* __restrict__ green_out,
    float* __restrict__ diff_out,
    int H, int W)
{
    __shared__ float tile[LDSH * LDSW];

    const int tid  = threadIdx.x;
    const int lane = tid & 31;
    const int wave = tid >> 5;           // 0..7
    const int hi   = lane >> 4;          // half selector
    const int lp   = lane & 15;          // pixel slot / M index

    const int x0 = blockIdx.x * TILE_W;
    const int y0 = blockIdx.y * TILE_H;
    const int b  = blockIdx.z;
    const size_t plane = (size_t)H * W;
    const float* img = mosaick + (size_t)b * plane;

    tile_fill(tile, img, x0, y0, H, W, tid);

    // per-lane A fragments: filter bank, constant across all tile rows
    const float* fb = (lp < 8) ? (sel_filts + lp * 25)
                               : (green_filts + (lp - 8) * 25);
    v2f afrag[7];
#pragma unroll
    for (int kk = 0; kk < 7; ++kk) {
        int t0 = 4 * kk + 2 * hi;
        int t1 = t0 + 1;
        afrag[kk].x = (t0 < 25) ? fb[t0] : 0.0f;
        afrag[kk].y = (t1 < 25) ? fb[t1] : 0.0f;
    }

    __syncthreads();

    // 64 wave-tasks: 4 column groups x 16 rows; 8 per wave, uniform per wave
    for (int task = wave; task < 64; task += 8) {
        const int cg = task & 3;
        const int r  = task >> 2;
        const int px = cg * 16 + lp;     // output x within tile

        v8f c = {};
#pragma unroll
        for (int kk = 0; kk < 7; ++kk) {
            int t0 = 4 * kk + 2 * hi;
            int t1 = t0 + 1;
            if (t0 >= 25) t0 = 0;        // padded K: A is zero, value ignored
            if (t1 >= 25) t1 = 0;
            v2f bfrag;
            bfrag.x = tile[(r + t0 / 5) * LDSW + px + (t0 % 5)];
            bfrag.y = tile[(r + t1 / 5) * LDSW + px + (t1 % 5)];
            c = __builtin_amdgcn_wmma_f32_16x16x4_f32(
                    false, afrag[kk], false, bfrag, (short)0, c, false, false);
        }

        // half-swap: lane p <-> lane p+16, so each lane holds both banks
        float partner[8];
#pragma unroll
        for (int k = 0; k < 8; ++k)
            partner[k] = __shfl(c[k], lane ^ 16, 32);

        float selv[8], gv[8];
#pragma unroll
        for (int k = 0; k < 8; ++k) {
            selv[k] = hi ? partner[k] : c[k];
            gv[k]   = hi ? c[k]       : partner[k];
        }

        // softmax(SIGMOID_PARAM * sel) dot green ; SIGMOID_PARAM == 1.0
        float mx = selv[0];
#pragma unroll
        for (int k = 1; k < 8; ++k) mx = fmaxf(mx, selv[k]);
        float se = 0.0f, sg = 0.0f;
#pragma unroll
        for (int k = 0; k < 8; ++k) {
            float e = __expf(selv[k] - mx);
            se += e;
            sg += e * gv[k];
        }
        float interp = sg / se;

        const int gx = x0 + px;
        const int gy = y0 + r;
        float mo   = tile[(r + HALO) * LDSW + (px + HALO)];
        bool  is_g = (((gy ^ gx) & 1) == 0);      // GRBG: G where parity matches
        float green = is_g ? mo : interp;
        float diff  = mo - green;

        if (hi == 0) {                            // lanes 0-15 own the 16 pixels
            size_t idx = (size_t)b * plane + (size_t)gy * W + gx;
            green_out[idx] = green;
            diff_out[idx]  = diff;
        }
    }
}

// ---------------------------------------------------------------------------
// Pass 2: three 5x5 convolutions of diff (h/v/q chroma) + R/B assembly.
// Only 75 FMA/pixel -> plain VALU with LDS halo tile.
// ---------------------------------------------------------------------------
__global__ __launch_bounds__(NT) void demosaic_pass2(
    const float* __restrict__ mosaick,
    const float* __restrict__ green_in,
    const float* __restrict__ diff_in,
    const float* __restrict__ hf,
    const float* __restrict__ vf,
    const float* __restrict__ qf,
    float* __restrict__ out,
    int H, int W)
{
    __shared__ float tile[LDSH * LDSW];
    __shared__ float f3[3][25];

    const int tid = threadIdx.x;
    const int x0 = blockIdx.x * TILE_W;
    const int y0 = blockIdx.y * TILE_H;
    const int b  = blockIdx.z;
    const size_t plane = (size_t)H * W;
    const float* dplane = diff_in + (size_t)b * plane;

    if (tid < 25) {
        f3[0][tid] = hf[tid];
        f3[1][tid] = vf[tid];
        f3[2][tid] = qf[tid];
    }

    tile_fill(tile, dplane, x0, y0, H, W, tid);
    __syncthreads();

    for (int p = tid; p < TILE_W * TILE_H; p += NT) {
        int r  = p / TILE_W;
        int px = p - r * TILE_W;
        float hc = 0.0f, vc = 0.0f, qc = 0.0f;
#pragma unroll
        for (int t = 0; t < 25; ++t) {
            float d = tile[(r + t / 5) * LDSW + px + (t % 5)];
            hc = fmaf(d, f3[0][t], hc);
            vc = fmaf(d, f3[1][t], vc);
            qc = fmaf(d, f3[2][t], qc);
        }
        const int gx = x0 + px;
        const int gy = y0 + r;
        size_t pidx = (size_t)gy * W + gx;
        float mo = mosaick[(size_t)b * plane + pidx];
        float g  = green_in[(size_t)b * plane + pidx];

        int  yod = gy & 1, xod = gx & 1;
        bool is_g      = (yod == xod);
        bool is_r      = (!yod) && xod;
        bool is_b      = yod && (!xod);
        bool g_redrow  = is_g && (!yod);
        bool g_bluerow = is_g && yod;

        float red  = is_r ? mo : (g_redrow  ? g + hc : (g_bluerow ? g + vc : g + qc));
        float blue = is_b ? mo : (g_bluerow ? g + hc : (g_redrow  ? g + vc : g + qc));

        float* ob = out + (size_t)b * 3 * plane;
        ob[0 * plane + pidx] = red;
        ob[1 * plane + pidx] = g;
        ob[2 * plane + pidx] = blue;
    }
}

extern "C" void kernel_launch(void* const* d_in, const int* in_sizes, int n_in,
                              void* d_out, int out_size, void* d_ws, size_t ws_size,
                              hipStream_t stream)
{
    const float* mosaick = (const float*)d_in[0];   // [B,1,H,W]
    const float* sel     = (const float*)d_in[1];   // [8,5,5]
    const float* grn     = (const float*)d_in[2];   // [8,5,5]
    const float* hf      = (const float*)d_in[3];   // [5,5]
    const float* vf      = (const float*)d_in[4];   // [5,5]
    const float* qf      = (const float*)d_in[5];   // [5,5]

    const int H = 1024, W = 1024;
    const int B = in_sizes[0] / (H * W);
    const size_t plane = (size_t)H * W;

    float* green = (float*)d_ws;                    // B*H*W f32
    float* diff  = green + (size_t)B * plane;       // B*H*W f32

    dim3 grid(W / TILE_W, H / TILE_H, B);
    demosaic_pass1<<<grid, NT, 0, stream>>>(mosaick, sel, grn, green, diff, H, W);
    demosaic_pass2<<<grid, NT, 0, stream>>>(mosaick, green, diff, hf, vf, qf,
                                            (float*)d_out, H, W);
}